// ContextHyper_58858231824542
// MI455X (gfx1250) — compile-verified
//
#include <hip/hip_runtime.h>
#include <math.h>

#define NA    2048
#define EF    128
#define NVIEW 4
#define NP    13
#define NGRP  8
#define HH    256
#define WW    352
#define NTOT  65536
#define WCOLS 416   // V * L * P * G

typedef __attribute__((ext_vector_type(2))) float v2f;
typedef __attribute__((ext_vector_type(8))) float v8f;

__device__ __forceinline__ float wsum32(float v) {
#pragma unroll
  for (int m = 1; m < 32; m <<= 1) v += __shfl_xor(v, m, 32);
  return v;
}
__device__ __forceinline__ float hsum16(float v) {
#pragma unroll
  for (int m = 1; m < 16; m <<= 1) v += __shfl_xor(v, m, 32);
  return v;
}

// ---------------- precompute: wbar = rowmean(op_w); A=wbar*g; C_g=sum wbar*beta; D_g=sum A; bbar=mean(op_b)
__global__ __launch_bounds__(128) void k_pre(
    const float* __restrict__ op_w, const float* __restrict__ op_b,
    const float* __restrict__ enc_g, const float* __restrict__ enc_beta,
    float* __restrict__ Aarr, float* __restrict__ Carr,
    float* __restrict__ Darr, float* __restrict__ bbar)
{
  __shared__ float sA[128], sWB[128], sB[128];
  int e = threadIdx.x;
  float wb = 0.f;
  for (int j = 0; j < 128; ++j) wb += op_w[e * 128 + j];
  wb *= (1.f / 128.f);
  float a = wb * enc_g[e];
  Aarr[e] = a;
  sA[e] = a;
  sWB[e] = wb * enc_beta[e];
  sB[e] = op_b[e];
  __syncthreads();
  if (e < 8) {
    float cc = 0.f, dd = 0.f;
    for (int t = 0; t < 16; ++t) { cc += sWB[e * 16 + t]; dd += sA[e * 16 + t]; }
    Carr[e] = cc; Darr[e] = dd;
  }
  if (e == 0) {
    float s = 0.f;
    for (int t = 0; t < 128; ++t) s += sB[t];
    bbar[0] = s * (1.f / 128.f);
  }
}

// ---------------- inst[n][e] = LN(relu(score*w+b))*g+beta, one wave per anchor
__global__ __launch_bounds__(256) void k_inst(
    const float* __restrict__ gsc, const int* __restrict__ aidx,
    const float* __restrict__ enc_w, const float* __restrict__ enc_b,
    const float* __restrict__ enc_g, const float* __restrict__ enc_beta,
    float* __restrict__ inst)
{
  const int wv = threadIdx.x >> 5;
  const int lane = threadIdx.x & 31;
  const int n = blockIdx.x * 8 + wv;
  float s = gsc[aidx[n]];
  int e0 = lane * 4;
  float x[4];
#pragma unroll
  for (int i = 0; i < 4; ++i) x[i] = fmaxf(fmaf(s, enc_w[e0 + i], enc_b[e0 + i]), 0.f);
  float sr = wsum32(x[0] + x[1] + x[2] + x[3]);
  float mu = sr * (1.f / 128.f);
  float sd = 0.f;
#pragma unroll
  for (int i = 0; i < 4; ++i) { float d = x[i] - mu; sd += d * d; }
  sd = wsum32(sd);
  float rstd = 1.f / sqrtf(sd * (1.f / 128.f) + 1e-5f);
#pragma unroll
  for (int i = 0; i < 4; ++i)
    inst[n * EF + e0 + i] = (x[i] - mu) * rstd * enc_g[e0 + i] + enc_beta[e0 + i];
}

// ---------------- wlog = inst @ wfc_w + wfc_b  via V_WMMA_F32_16X16X4_F32
// A 16x4 f32 frag: M = lane&15; VGPR0=K(k+2h), VGPR1=K(k+2h+1)   (h = lane>=16)
// B 4x16 f32 frag: N = lane&15; same K split
// C/D: elem r -> row m0 + 8h + r, col n0 + (lane&15)
__global__ __launch_bounds__(128) void k_gemm(
    const float* __restrict__ inst, const float* __restrict__ wfc_w,
    const float* __restrict__ wfc_b, float* __restrict__ wlog)
{
  const int wv = threadIdx.x >> 5;
  const int lane = threadIdx.x & 31;
  const int tile = blockIdx.x * 4 + wv;
  const int mt = tile / 26, nt = tile % 26;
  const int m0 = mt * 16, n0 = nt * 16;
  const int l16 = lane & 15;
  const int h = lane >> 4;

  v8f c = {0.f, 0.f, 0.f, 0.f, 0.f, 0.f, 0.f, 0.f};
  const float* arow = inst + (m0 + l16) * EF + 2 * h;
  const float* bcol = wfc_w + (2 * h) * WCOLS + (n0 + l16);
#pragma unroll 8
  for (int k = 0; k < 128; k += 4) {
    v2f a = *(const v2f*)(arow + k);
    v2f b;
    b.x = bcol[k * WCOLS];
    b.y = bcol[(k + 1) * WCOLS];
    c = __builtin_amdgcn_wmma_f32_16x16x4_f32(false, a, false, b, (short)0, c, false, false);
  }
  float bias = wfc_b[n0 + l16];
  int rowb = m0 + h * 8;
#pragma unroll
  for (int r = 0; r < 8; ++r)
    wlog[(rowb + r) * WCOLS + (n0 + l16)] = c[r] + bias;
}

// ---------------- main per-anchor kernel: one wave per anchor
__global__ __launch_bounds__(256) void k_main(
    const float* __restrict__ means, const float* __restrict__ scales,
    const float* __restrict__ rots, const float* __restrict__ smaps,
    const float* __restrict__ proj, const float* __restrict__ wh,
    const int* __restrict__ aidx,
    const float* __restrict__ enc_w, const float* __restrict__ enc_b,
    const float* __restrict__ lfc_w, const float* __restrict__ lfc_b,
    const float* __restrict__ inst, const float* __restrict__ wlog,
    const float* __restrict__ Aarr, const float* __restrict__ Carr,
    const float* __restrict__ Darr, const float* __restrict__ bbar,
    float* __restrict__ si)
{
  __shared__ float sp_s[8][NP][3];
  const int wv = threadIdx.x >> 5;
  const int lane = threadIdx.x & 31;
  const int n = blockIdx.x * 8 + wv;
  const int h = lane >> 4;
  const int g0 = aidx[n];

  float wE[4], bE[4], aE[4], iE[4];
#pragma unroll
  for (int i = 0; i < 4; ++i) {
    int e = lane + 32 * i;
    wE[i] = enc_w[e]; bE[i] = enc_b[e]; aE[i] = Aarr[e];
    iE[i] = inst[n * EF + e];
  }
  float Cg[4], Dg[4];
#pragma unroll
  for (int i = 0; i < 4; ++i) { int g = 2 * i + h; Cg[i] = Carr[g]; Dg[i] = Darr[g]; }

  // learned points: sigmoid(inst @ lfc_w + lfc_b) - 0.5
  float learn[18];
#pragma unroll
  for (int j = 0; j < 18; ++j) {
    float p = 0.f;
#pragma unroll
    for (int i = 0; i < 4; ++i) p = fmaf(iE[i], lfc_w[(lane + 32 * i) * 18 + j], p);
    p = wsum32(p) + lfc_b[j];
    learn[j] = 1.f / (1.f + __expf(-p)) - 0.5f;
  }

  // quaternion -> rotation (reference ordering)
  float q0 = rots[g0 * 4 + 0], q1 = rots[g0 * 4 + 1], q2 = rots[g0 * 4 + 2], q3 = rots[g0 * 4 + 3];
  float inr = 1.f / sqrtf(q0 * q0 + q1 * q1 + q2 * q2 + q3 * q3);
  float qw = q0 * inr, qx = q1 * inr, qy = q2 * inr, qz = q3 * inr;
  float R00 = 1.f - 2.f * (qy * qy + qz * qz), R01 = 2.f * (qx * qy - qw * qz), R02 = 2.f * (qx * qz + qw * qy);
  float R10 = 2.f * (qx * qy + qw * qz), R11 = 1.f - 2.f * (qx * qx + qz * qz), R12 = 2.f * (qy * qz - qw * qx);
  float R20 = 2.f * (qx * qz - qw * qy), R21 = 2.f * (qy * qz + qw * qx), R22 = 1.f - 2.f * (qx * qx + qy * qy);
  float m0 = means[g0 * 3], m1 = means[g0 * 3 + 1], m2 = means[g0 * 3 + 2];
  float s0 = scales[g0 * 3], s1 = scales[g0 * 3 + 1], s2 = scales[g0 * 3 + 2];

  // pre-scaled sample points into LDS (per-wave slice)
  if (lane == 0) {
    const float FXx[7] = {0.f, 0.45f, -0.45f, 0.f, 0.f, 0.f, 0.f};
    const float FXy[7] = {0.f, 0.f, 0.f, 0.45f, -0.45f, 0.f, 0.f};
    const float FXz[7] = {0.f, 0.f, 0.f, 0.f, 0.f, 0.45f, -0.45f};
#pragma unroll
    for (int p = 0; p < 7; ++p) {
      sp_s[wv][p][0] = FXx[p] * s0; sp_s[wv][p][1] = FXy[p] * s1; sp_s[wv][p][2] = FXz[p] * s2;
    }
#pragma unroll
    for (int p = 7; p < NP; ++p) {
      sp_s[wv][p][0] = learn[(p - 7) * 3 + 0] * s0;
      sp_s[wv][p][1] = learn[(p - 7) * 3 + 1] * s1;
      sp_s[wv][p][2] = learn[(p - 7) * 3 + 2] * s2;
    }
  }
  __syncthreads();

  // per-group softmax stats over the 52 (v,p) slots; lane owns group lane%8
  float wvv[13];
  float gmax = -3.402823466e+38f;
#pragma unroll
  for (int c = 0; c < 13; ++c) {
    wvv[c] = wlog[n * WCOLS + c * 32 + lane];
    gmax = fmaxf(gmax, wvv[c]);
  }
  gmax = fmaxf(gmax, __shfl_xor(gmax, 8, 32));
  gmax = fmaxf(gmax, __shfl_xor(gmax, 16, 32));
  float gden = 0.f;
#pragma unroll
  for (int c = 0; c < 13; ++c) gden += __expf(wvv[c] - gmax);
  gden += __shfl_xor(gden, 8, 32);
  gden += __shfl_xor(gden, 16, 32);
  float mg[4], dg[4];
#pragma unroll
  for (int i = 0; i < 4; ++i) {
    mg[i] = __shfl(gmax, 2 * i + h, 32);
    dg[i] = __shfl(gden, 2 * i + h, 32);
  }

  float acc = 0.f;
#pragma unroll
  for (int v = 0; v < NVIEW; ++v) {
    float P00 = proj[v * 16 + 0], P01 = proj[v * 16 + 1], P02 = proj[v * 16 + 2], P03 = proj[v * 16 + 3];
    float P10 = proj[v * 16 + 4], P11 = proj[v * 16 + 5], P12 = proj[v * 16 + 6], P13 = proj[v * 16 + 7];
    float P20 = proj[v * 16 + 8], P21 = proj[v * 16 + 9], P22 = proj[v * 16 + 10], P23 = proj[v * 16 + 11];
    float iwx = 1.f / wh[v * 2 + 0], iwy = 1.f / wh[v * 2 + 1];
    for (int p = 0; p < NP; ++p) {
      float a0 = sp_s[wv][p][0], a1 = sp_s[wv][p][1], a2 = sp_s[wv][p][2];
      float kx = m0 + R00 * a0 + R01 * a1 + R02 * a2;
      float ky = m1 + R10 * a0 + R11 * a1 + R12 * a2;
      float kz = m2 + R20 * a0 + R21 * a1 + R22 * a2;
      float pa = P00 * kx + P01 * ky + P02 * kz + P03;
      float pb = P10 * kx + P11 * ky + P12 * kz + P13;
      float pc = P20 * kx + P21 * ky + P22 * kz + P23;
      float zc = fmaxf(pc, 1e-5f);
      float u = pa / zc * iwx, vv2 = pb / zc * iwy;
      float x = u * (float)WW - 0.5f, y = vv2 * (float)HH - 0.5f;
      float x0f = floorf(x), y0f = floorf(y);
      float fx = x - x0f, fy = y - y0f;
      float sg[4] = {0.f, 0.f, 0.f, 0.f};
#pragma unroll
      for (int c = 0; c < 4; ++c) {
        float cx = x0f + (float)(c & 1);
        float cy = y0f + (float)(c >> 1);
        float cw = ((c & 1) ? fx : 1.f - fx) * ((c >> 1) ? fy : 1.f - fy);
        bool valid = (cx >= 0.f) && (cx <= (float)(WW - 1)) && (cy >= 0.f) && (cy <= (float)(HH - 1));
        float f = valid ? cw : 0.f;   // uniform across wave
        if (f != 0.f) {
          int ix = (int)cx, iy = (int)cy;
          float s = smaps[v * (HH * WW) + iy * WW + ix];
          float r[4], t[4];
          float sr = 0.f;
#pragma unroll
          for (int i = 0; i < 4; ++i) {
            r[i] = fmaxf(fmaf(s, wE[i], bE[i]), 0.f);
            sr += r[i];
          }
          sr = wsum32(sr);
          float mu = sr * (1.f / 128.f);
          float sd = 0.f;
#pragma unroll
          for (int i = 0; i < 4; ++i) { float d = r[i] - mu; sd += d * d; t[i] = aE[i] * r[i]; }
          sd = wsum32(sd);
          float rstd = 1.f / sqrtf(sd * (1.f / 128.f) + 1e-5f);
#pragma unroll
          for (int i = 0; i < 4; ++i) {
            float S = hsum16(t[i]);  // group sum (16-lane half), group = 2i+h
            sg[i] += f * (rstd * (S - mu * Dg[i]) + Cg[i]);
          }
        }
      }
      float contrib = 0.f;
      int base = n * WCOLS + (v * NP + p) * NGRP;
#pragma unroll
      for (int i = 0; i < 4; ++i) {
        float l = wlog[base + 2 * i + h];
        contrib += (__expf(l - mg[i]) / dg[i]) * sg[i];
      }
      if ((lane & 15) == 0) acc += contrib;  // lanes 0 & 16 carry the 8 groups
    }
  }
  acc = wsum32(acc);
  if (lane == 0) si[n] = acc + bbar[0];
}

// ---------------- min/max over scores + bitonic sort of si + quantile blend
__global__ __launch_bounds__(1024) void k_final(
    const float* __restrict__ gsc, const float* __restrict__ si,
    float* __restrict__ out2)
{
  __shared__ float sdat[2048];
  __shared__ float sred[1024];
  __shared__ float s_mn, s_mx;
  const int t = threadIdx.x;
  float lmin = 3.402823466e+38f, lmax = -3.402823466e+38f;
  for (int i = t; i < NTOT; i += 1024) {
    float v = gsc[i];
    lmin = fminf(lmin, v); lmax = fmaxf(lmax, v);
  }
  sred[t] = lmin; __syncthreads();
  for (int s = 512; s > 0; s >>= 1) { if (t < s) sred[t] = fminf(sred[t], sred[t + s]); __syncthreads(); }
  if (t == 0) s_mn = sred[0];
  __syncthreads();
  sred[t] = lmax; __syncthreads();
  for (int s = 512; s > 0; s >>= 1) { if (t < s) sred[t] = fmaxf(sred[t], sred[t + s]); __syncthreads(); }
  if (t == 0) s_mx = sred[0];

  sdat[t] = si[t];
  sdat[t + 1024] = si[t + 1024];
  __syncthreads();
  for (int k = 2; k <= 2048; k <<= 1) {
    for (int j = k >> 1; j > 0; j >>= 1) {
#pragma unroll
      for (int r = 0; r < 2; ++r) {
        int i = t + r * 1024;
        int l = i ^ j;
        if (l > i) {
          float a = sdat[i], b = sdat[l];
          bool asc = ((i & k) == 0);
          if ((a > b) == asc) { sdat[i] = b; sdat[l] = a; }
        }
      }
      __syncthreads();
    }
  }
  if (t == 0) {
    float mn = s_mn, mx = s_mx;
    float pl = 0.05f * 2047.f;
    int il = (int)pl; float fl = pl - (float)il;
    float vl = sdat[il] + fl * (sdat[il + 1] - sdat[il]);
    float ph = 0.95f * 2047.f;
    int ih = (int)ph; float fh = ph - (float)ih;
    float vh = sdat[ih] + fh * (sdat[ih + 1] - sdat[ih]);
    out2[0] = mn + (1.f / (1.f + expf(-vl))) * (mx - mn);
    out2[1] = mn + (1.f / (1.f + expf(-vh))) * (mx - mn);
  }
}

extern "C" void kernel_launch(void* const* d_in, const int* in_sizes, int n_in,
                              void* d_out, int out_size, void* d_ws, size_t ws_size,
                              hipStream_t stream) {
  (void)in_sizes; (void)n_in; (void)out_size; (void)ws_size;
  const float* means    = (const float*)d_in[0];
  const float* scales   = (const float*)d_in[1];
  const float* rots     = (const float*)d_in[2];
  /* d_in[3] = opacities: unused by the reference */
  const float* gsc      = (const float*)d_in[4];
  const float* smaps    = (const float*)d_in[5];
  const float* proj     = (const float*)d_in[6];
  const float* wh       = (const float*)d_in[7];
  const int*   aidx     = (const int*)d_in[8];
  const float* enc_w    = (const float*)d_in[9];
  const float* enc_b    = (const float*)d_in[10];
  const float* enc_g    = (const float*)d_in[11];
  const float* enc_beta = (const float*)d_in[12];
  const float* lfc_w    = (const float*)d_in[13];
  const float* lfc_b    = (const float*)d_in[14];
  const float* wfc_w    = (const float*)d_in[15];
  const float* wfc_b    = (const float*)d_in[16];
  const float* op_w     = (const float*)d_in[17];
  const float* op_b     = (const float*)d_in[18];

  float* ws   = (float*)d_ws;
  float* inst = ws;                    // 2048*128
  float* wlog = inst + NA * EF;        // 2048*416
  float* sia  = wlog + NA * WCOLS;     // 2048
  float* Aarr = sia + NA;              // 128
  float* Carr = Aarr + EF;             // 8
  float* Darr = Carr + NGRP;           // 8
  float* bbar = Darr + NGRP;           // 1

  k_pre<<<1, 128, 0, stream>>>(op_w, op_b, enc_g, enc_beta, Aarr, Carr, Darr, bbar);
  k_inst<<<NA / 8, 256, 0, stream>>>(gsc, aidx, enc_w, enc_b, enc_g, enc_beta, inst);
  k_gemm<<<(NA / 16) * (WCOLS / 16) / 4, 128, 0, stream>>>(inst, wfc_w, wfc_b, wlog);
  k_main<<<NA / 8, 256, 0, stream>>>(means, scales, rots, smaps, proj, wh, aidx,
                                     enc_w, enc_b, lfc_w, lfc_b, inst, wlog,
                                     Aarr, Carr, Darr, bbar, sia);
  k_final<<<1, 1024, 0, stream>>>(gsc, sia, (float*)d_out);
}